// GHM_Loss_70677981823512
// MI455X (gfx1250) — compile-verified
//
#include <hip/hip_runtime.h>
#include <stdint.h>

// GHM loss (focal + GHM-R), single-pass. CDNA5 path: double-buffered
// global_load_async_to_lds_b128 staging + s_wait_asynccnt pipelining,
// register histograms, ds_add_f64 / global_atomic_add_f64 reductions.

#define TPB     256
#define MAXBLK  4096
#define NBINS   10
#define ALPHA_C 0.25f
#define EPS_C   1e-05f
#define MU_C    0.02f

// d_ws layout (doubles): [0..9] bin counts, [10..19] bin loss sums,
//                        [20] focal sum, [21] valid-pixel count
#define WS_N 22

__global__ void ghm_init_kernel(double* __restrict__ ws) {
    int t = threadIdx.x;
    if (t < WS_N) ws[t] = 0.0;
}

__global__ __launch_bounds__(TPB) void ghm_main_kernel(
    const float* __restrict__ preds,
    const float* __restrict__ tgts,
    double* __restrict__ ws,
    int npix, int stride_pix)
{
    // Staging: [buffer][chunk][tid][4 floats]. chunk 0/1 = preds lo/hi,
    // chunk 2/3 = targets lo/hi. 32KB total.
    __shared__ __align__(16) float tile[2][4][TPB][4];
    __shared__ double red[WS_N];

    const int tid = threadIdx.x;
    if (tid < WS_N) red[tid] = 0.0;
    __syncthreads();

    const int base  = (int)(blockIdx.x) * TPB + tid;
    // Uniform per-block iteration count (thread 0 of the block bounds it).
    const int iters = (npix - 1 - (int)(blockIdx.x) * TPB) / stride_pix + 1;

    float cnt[NBINS], lsum[NBINS];
#pragma unroll
    for (int b = 0; b < NBINS; ++b) { cnt[b] = 0.0f; lsum[b] = 0.0f; }
    float facc = 0.0f;   // focal loss partial
    float vacc = 0.0f;   // valid pixel count partial

    // Issue 4 async 16B global->LDS copies for pixel i into buffer `buf`.
    // Address is clamped so every wave always issues exactly 4 ops per stage
    // -> ASYNCcnt watermark arithmetic below is exact (in-order completion).
    auto stage = [&](int buf, int i) {
        int ic = i < npix ? i : (npix - 1);
        uint32_t go  = (uint32_t)ic * 32u;  // 8 floats per pixel
        uint32_t lp0 = (uint32_t)(uintptr_t)&tile[buf][0][tid][0];
        uint32_t lp1 = (uint32_t)(uintptr_t)&tile[buf][1][tid][0];
        uint32_t lt0 = (uint32_t)(uintptr_t)&tile[buf][2][tid][0];
        uint32_t lt1 = (uint32_t)(uintptr_t)&tile[buf][3][tid][0];
        asm volatile("global_load_async_to_lds_b128 %0, %1, %2 offset:0"
                     :: "v"(lp0), "v"(go), "s"(preds) : "memory");
        asm volatile("global_load_async_to_lds_b128 %0, %1, %2 offset:16"
                     :: "v"(lp1), "v"(go), "s"(preds) : "memory");
        asm volatile("global_load_async_to_lds_b128 %0, %1, %2 offset:0"
                     :: "v"(lt0), "v"(go), "s"(tgts) : "memory");
        asm volatile("global_load_async_to_lds_b128 %0, %1, %2 offset:16"
                     :: "v"(lt1), "v"(go), "s"(tgts) : "memory");
    };

    stage(0, base);  // prologue: fill buffer 0

    for (int k = 0; k < iters; ++k) {
        const int i = base + k * stride_pix;

        if (k + 1 < iters) {
            stage((k + 1) & 1, base + (k + 1) * stride_pix);
            // 8 outstanding; <=4 means the current buffer's 4 are done.
            asm volatile("s_wait_asynccnt 0x4" ::: "memory");
        } else {
            asm volatile("s_wait_asynccnt 0x0" ::: "memory");
        }

        const int buf = k & 1;
        float4 p0 = *(const float4*)&tile[buf][0][tid][0];
        float4 p1 = *(const float4*)&tile[buf][1][tid][0];
        float4 t0 = *(const float4*)&tile[buf][2][tid][0];
        float4 t1 = *(const float4*)&tile[buf][3][tid][0];

        const bool ok = (i < npix);

        // ---- focal loss on channel 0 ----
        float x  = p0.x, y = t0.x;
        float ty = 2.0f * y - 1.0f;
        float xt = x * ty + (1.0f - y);
        float at = ALPHA_C * ty + (1.0f - y);
        float om = 1.0f - xt;
        float fl = -at * om * om * logf(xt + EPS_C);
        facc += ok ? fl : 0.0f;

        const float vflag = (ok && (y > 0.1f)) ? 1.0f : 0.0f;
        vacc += vflag;

        // ---- GHM-R on channels 1..7 ----
        float pl[7] = {p0.y, p0.z, p0.w, p1.x, p1.y, p1.z, p1.w};
        float tl[7] = {t0.y, t0.z, t0.w, t1.x, t1.y, t1.z, t1.w};
#pragma unroll
        for (int c = 0; c < 7; ++c) {
            float d    = pl[c] - tl[c];
            float s2   = fmaf(d, d, MU_C * MU_C);
            float s    = sqrtf(s2);
            float loss = s - MU_C;
            float g    = fabsf(d) / s;          // in [0,1)
            int   bi   = (int)(g * (float)NBINS);
            bi = bi > (NBINS - 1) ? (NBINS - 1) : bi;
#pragma unroll
            for (int b = 0; b < NBINS; ++b) {
                float hit = (bi == b) ? vflag : 0.0f;
                cnt[b]  += hit;
                lsum[b]  = fmaf(hit, loss, lsum[b]);
            }
        }
    }

    // ---- block reduction (f64 in LDS), then one global f64 atomic per value
#pragma unroll
    for (int b = 0; b < NBINS; ++b) {
        atomicAdd(&red[b],          (double)cnt[b]);
        atomicAdd(&red[NBINS + b],  (double)lsum[b]);
    }
    atomicAdd(&red[20], (double)facc);
    atomicAdd(&red[21], (double)vacc);
    __syncthreads();
    if (tid < WS_N) unsafeAtomicAdd(&ws[tid], red[tid]);
}

__global__ void ghm_final_kernel(const double* __restrict__ ws,
                                 float* __restrict__ out, int npix)
{
    if (threadIdx.x != 0 || blockIdx.x != 0) return;
    double tot = ws[21];
    if (tot < 1.0) tot = 1.0;
    int n = 0;
    double reg = 0.0;
    for (int b = 0; b < NBINS; ++b) {
        double cb = ws[b];
        if (cb > 0.0) {
            ++n;
            double acc = 0.3 * cb;              // (1 - MMT) * counts
            if (acc < 1e-30) acc = 1e-30;
            reg += (tot / acc) * ws[NBINS + b]; // w_per_bin[b] * loss_sum[b]
        }
    }
    if (n > 0) reg /= (double)n;
    reg /= tot;
    double cls = ws[20] / (double)npix;
    out[0] = (float)(cls + reg);
    out[1] = (float)reg;
    out[2] = (float)cls;
}

extern "C" void kernel_launch(void* const* d_in, const int* in_sizes, int n_in,
                              void* d_out, int out_size, void* d_ws, size_t ws_size,
                              hipStream_t stream) {
    const float* preds = (const float*)d_in[0];
    const float* tgts  = (const float*)d_in[1];
    double* ws  = (double*)d_ws;
    float*  out = (float*)d_out;

    const int npix = in_sizes[0] / 8;           // 32*400*352 pixels, 8 ch each
    int nblk = (npix + TPB - 1) / TPB;
    if (nblk > MAXBLK) nblk = MAXBLK;
    const int stride_pix = nblk * TPB;

    hipLaunchKernelGGL(ghm_init_kernel,  dim3(1),    dim3(32),  0, stream, ws);
    hipLaunchKernelGGL(ghm_main_kernel,  dim3(nblk), dim3(TPB), 0, stream,
                       preds, tgts, ws, npix, stride_pix);
    hipLaunchKernelGGL(ghm_final_kernel, dim3(1),    dim3(1),   0, stream,
                       ws, out, npix);
}